// GMM_64158221467827
// MI455X (gfx1250) — compile-verified
//
#include <hip/hip_runtime.h>
#include <math.h>

// D = 2 (padded to K=4 for V_WMMA_F32_16X16X4_F32), VAR = 1.0
typedef __attribute__((ext_vector_type(2))) float v2f;
typedef __attribute__((ext_vector_type(8))) float v8f;

#define ROWS_PER_WAVE   16
#define WAVES_PER_BLOCK 8
#define ROWS_PER_BLOCK  (ROWS_PER_WAVE * WAVES_PER_BLOCK)

__global__ __launch_bounds__(256) void gmm_wmma_kernel(
    const float* __restrict__ x,        // [N,2]
    const float* __restrict__ means,    // [M,2]
    const float* __restrict__ weights,  // [M]
    float* __restrict__ out,            // [N]
    int N, int M)
{
  const int lane = threadIdx.x & 31;
  const int wave = threadIdx.x >> 5;
  const int l16  = lane & 15;
  const int half = lane >> 4;           // 0: K=0/1 lanes, 1: K=2/3 (pad) lanes

  const int rowBase = (blockIdx.x * WAVES_PER_BLOCK + wave) * ROWS_PER_WAVE;

  // ---- A fragment: 16x4 f32. Lanes 0-15: row=l16, K=0/1 = (x,y).
  // Lanes 16-31 hold K=2/3 pad slots; their values are don't-care because
  // B's K=2/3 entries are forced to zero below.
  int r  = rowBase + l16;
  int rc = r < N ? r : (N - 1);
  const float xx = x[2 * rc + 0];
  const float xy = x[2 * rc + 1];
  v2f a; a[0] = xx; a[1] = xy;

  // ||x_row||^2 for row = l16 (identical in both halves of the wave)
  const float myx2 = xx * xx + xy * xy;

  // Accumulator slot v of this lane corresponds to row (v + half*8);
  // fetch that row's ||x||^2 via an in-wave gather.
  float x2r[8];
  #pragma unroll
  for (int v = 0; v < 8; ++v)
    x2r[v] = __shfl(myx2, v + half * 8, 32);

  const float bsel = (half == 0) ? 1.0f : 0.0f;   // zero the K=2/3 pad in B

  float acc[8];
  #pragma unroll
  for (int v = 0; v < 8; ++v) acc[v] = 0.0f;

  const int ngroups = M >> 4;           // 16 components per WMMA column tile
  #pragma unroll 4
  for (int g = 0; g < ngroups; ++g) {
    const int n = (g << 4) + l16;       // this lane's component (column) index
    const float mx = means[2 * n + 0];
    const float my = means[2 * n + 1];
    const float m2 = mx * mx + my * my;
    const float wn = weights[n];

    v2f b; b[0] = mx * bsel; b[1] = my * bsel;

    v8f c = {};
    // D = A(16x4) x B(4x16): 256 dot products x·m in one instruction
    v8f dot = __builtin_amdgcn_wmma_f32_16x16x4_f32(
        /*neg_a=*/false, a, /*neg_b=*/false, b,
        /*c_mod=*/(short)0, c, /*reuse_a=*/false, /*reuse_b=*/false);

    #pragma unroll
    for (int v = 0; v < 8; ++v) {
      const float d2 = x2r[v] + m2 - 2.0f * dot[v];   // ||x-m||^2
      acc[v] += wn * __expf(-0.5f * d2);
    }
  }

  // Reduce over the 16 columns held by each half-wave
  // (masks 1,2,4,8 never cross the 16-lane halves).
  #pragma unroll
  for (int v = 0; v < 8; ++v) {
    float s = acc[v];
    s += __shfl_xor(s, 1, 32);
    s += __shfl_xor(s, 2, 32);
    s += __shfl_xor(s, 4, 32);
    s += __shfl_xor(s, 8, 32);
    acc[v] = s;
  }

  const float coef = 0.15915494309189535f;  // 1 / (2*pi*VAR)
  if (l16 == 0) {                            // lane 0 -> rows 0..7, lane 16 -> rows 8..15
    #pragma unroll
    for (int v = 0; v < 8; ++v) {
      const int row = rowBase + half * 8 + v;
      if (row < N) out[row] = coef * acc[v];
    }
  }
}

extern "C" void kernel_launch(void* const* d_in, const int* in_sizes, int n_in,
                              void* d_out, int out_size, void* d_ws, size_t ws_size,
                              hipStream_t stream) {
  const float* x       = (const float*)d_in[0];
  const float* means   = (const float*)d_in[1];
  const float* weights = (const float*)d_in[2];
  float* out = (float*)d_out;

  const int N = in_sizes[0] / 2;   // x is [N,2]
  const int M = in_sizes[2];       // weights is [M]; M=256, multiple of 16

  const int blocks = (N + ROWS_PER_BLOCK - 1) / ROWS_PER_BLOCK;
  gmm_wmma_kernel<<<blocks, 256, 0, stream>>>(x, means, weights, out, N, M);
}